// NISQSIMClassifier_75694503624953
// MI455X (gfx1250) — compile-verified
//
#include <hip/hip_runtime.h>
#include <hip/hip_bf16.h>
#include <math.h>

// ---------------------------------------------------------------------------
// NISQ classifier on MI455X (gfx1250), Pauli-transfer-matrix formulation.
// r[P] = tr(P rho), P in {I,X,Y,Z}^10  ->  real f32 vector of 4^10 entries.
// Every gate/noise channel = real 16x16 matrix on a pair of radix-4 axes,
// applied with V_WMMA_F32_16X16X4_F32 (four K=4 accumulating WMMAs per tile).
// ---------------------------------------------------------------------------

#define NQ 10
#define NL 3
#define DIMN 1024
#define BATCH 2048
#define RLEN (1u << 20)     // 4^10 Pauli coefficients
#define NCOLS 65536         // 4^8 free columns for any axis pair
#define MAT_STRIDE 256      // one 16x16 matrix

typedef float v2f __attribute__((ext_vector_type(2)));
typedef float v8f __attribute__((ext_vector_type(8)));

// ---------------- small dense helpers (prep kernel only) -------------------
__device__ __forceinline__ void mat3mul(const float* A, const float* B, float* C) {
  for (int i = 0; i < 3; ++i)
    for (int j = 0; j < 3; ++j) {
      float s = 0.f;
      for (int k = 0; k < 3; ++k) s += A[i * 3 + k] * B[k * 3 + j];
      C[i * 3 + j] = s;
    }
}
__device__ __forceinline__ void mat4mul(const float* A, const float* B, float* C) {
  for (int i = 0; i < 4; ++i)
    for (int j = 0; j < 4; ++j) {
      float s = 0.f;
      for (int k = 0; k < 4; ++k) s += A[i * 4 + k] * B[k * 4 + j];
      C[i * 4 + j] = s;
    }
}
// K16[(4i+j)*16 + (4k+l)] = A[i][k] * B[j][l]   (A acts on the high axis)
__device__ __forceinline__ void kron44(const float* A, const float* B, float* K16) {
  for (int i = 0; i < 4; ++i)
    for (int j = 0; j < 4; ++j)
      for (int k = 0; k < 4; ++k)
        for (int l = 0; l < 4; ++l)
          K16[(4 * i + j) * 16 + (4 * k + l)] = A[i * 4 + k] * B[j * 4 + l];
}
// Pauli-transfer matrix of phase∘amp∘depol(p)  (applied left-to-right order:
// depol first, phase last -> N = Rp * Ra * Rd)
__device__ __forceinline__ void noise1(float* N, float pdep) {
  const float gA = (float)(1.0 - exp(-1.0e-7 / 0.05));  // amplitude damping
  const float gP = (float)(1.0 - exp(-1.0e-7 / 0.07));  // phase damping
  const float sA = sqrtf(1.f - gA), sP = sqrtf(1.f - gP);
  const float lam = 1.f - 4.f * pdep / 3.f;
  float Rd[16] = {1, 0, 0, 0, 0, lam, 0, 0, 0, 0, lam, 0, 0, 0, 0, lam};
  float Ra[16] = {1, 0, 0, 0, 0, sA, 0, 0, 0, 0, sA, 0, gA, 0, 0, 1.f - gA};
  float Rp[16] = {1, 0, 0, 0, 0, sP, 0, 0, 0, 0, sP, 0, 0, 0, 0, 1};
  float t[16];
  mat4mul(Ra, Rd, t);
  mat4mul(Rp, t, N);
}
// PTM of qml.Rot = RZ(om) RY(th) RZ(ph): Bloch rotation Rz(om)*Ry(th)*Rz(ph)
__device__ __forceinline__ void rotPTM(float* M, float ph, float th, float om) {
  float R1[9] = {cosf(ph), -sinf(ph), 0, sinf(ph), cosf(ph), 0, 0, 0, 1};
  float R2[9] = {cosf(th), 0, sinf(th), 0, 1, 0, -sinf(th), 0, cosf(th)};
  float R3[9] = {cosf(om), -sinf(om), 0, sinf(om), cosf(om), 0, 0, 0, 1};
  float t[9], r[9];
  mat3mul(R2, R1, t);
  mat3mul(R3, t, r);
  M[0] = 1.f; M[1] = 0.f; M[2] = 0.f; M[3] = 0.f;
  for (int i = 0; i < 3; ++i) {
    M[(i + 1) * 4] = 0.f;
    for (int j = 0; j < 3; ++j) M[(i + 1) * 4 + j + 1] = r[i * 3 + j];
  }
}

// ------------- build all 16x16 pair matrices (tiny, one block) -------------
// slots 0..14 : layer l, pair j -> (noise*Rot)_q ⊗ (noise*Rot)_{q+1}, q=2j
// slot 15     : (N2 ⊗ N2) * PTM(CNOT)   (identical for every CNOT in circuit)
// slot 16     : bitflip ⊗ bitflip (readout)
__global__ void prep_mats(const float* __restrict__ cw, float* __restrict__ mats) {
  const int t = threadIdx.x;
  if (t < NL * 5) {
    const int l = t / 5, j = t % 5, q = 2 * j;
    float N[16], Rr[16], Ma[16], Mb[16];
    noise1(N, 0.001f);
    rotPTM(Rr, cw[(l * NQ + q) * 3 + 0], cw[(l * NQ + q) * 3 + 1], cw[(l * NQ + q) * 3 + 2]);
    mat4mul(N, Rr, Ma);
    rotPTM(Rr, cw[(l * NQ + q + 1) * 3 + 0], cw[(l * NQ + q + 1) * 3 + 1],
           cw[(l * NQ + q + 1) * 3 + 2]);
    mat4mul(N, Rr, Mb);
    kron44(Ma, Mb, mats + t * MAT_STRIDE);
  } else if (t == NL * 5) {
    float N[16];
    noise1(N, 0.01f);
    float K[256];
    kron44(N, N, K);
    // CNOT Pauli map (control ⊗ target): input pair jin -> output pair, sign
    const int oidx[16] = {0, 1, 14, 15, 5, 4, 11, 10, 9, 8, 7, 6, 12, 13, 2, 3};
    const float sg[16] = {1, 1, 1, 1, 1, 1, 1, -1, 1, 1, -1, 1, 1, 1, 1, 1};
    float* C2 = mats + 15 * MAT_STRIDE;
    for (int i = 0; i < 16; ++i)
      for (int jn = 0; jn < 16; ++jn) C2[i * 16 + jn] = sg[jn] * K[i * 16 + oidx[jn]];
  } else if (t == NL * 5 + 1) {
    const float bf = 1.f - 2.f * 0.02f;  // bitflip: diag(1,1,1-2p,1-2p)
    float B[16] = {1, 0, 0, 0, 0, 1, 0, 0, 0, 0, bf, 0, 0, 0, 0, bf};
    kron44(B, B, mats + 16 * MAT_STRIDE);
  }
}

// |0..0><0..0|  ->  r[idx] = 1 iff every base-4 digit is I(0) or Z(3)
__global__ void init_r(float* __restrict__ r) {
  const unsigned idx = blockIdx.x * blockDim.x + threadIdx.x;
  if (idx < RLEN) r[idx] = (((idx ^ (idx >> 1)) & 0x55555u) == 0u) ? 1.f : 0.f;
}

// ---------------------------------------------------------------------------
// Apply a real 16x16 matrix along an axis pair of r, in place, with WMMA.
// Element i = 4a+b lives at base + a*sFirst + b*sSecond.  Column c of the
// remaining 8 digits expands to base via (lgL, lgM) digit re-insertion.
// ---------------------------------------------------------------------------
__global__ void __launch_bounds__(256) apply_pair(
    float* __restrict__ r, const float* __restrict__ M16,
    int lgL, int lgM, int sFirst, int sSecond) {
  const int lane = threadIdx.x & 31;
  const int wv = blockIdx.x * (blockDim.x >> 5) + (threadIdx.x >> 5);
  const int nWaves = gridDim.x * (blockDim.x >> 5);
  const int row = lane & 15;
  const int hi = lane >> 4;
  const int k0 = 2 * hi;

  // A fragments: ISA layout for 32-bit 16x4 A — lane holds M=row,
  // VGPR0 = K {0|2}, VGPR1 = K {1|3} (per chunk of 4 K-columns).
  v2f A0, A1, A2, A3;
  A0.x = M16[row * 16 + 0 + k0];  A0.y = M16[row * 16 + 1 + k0];
  A1.x = M16[row * 16 + 4 + k0];  A1.y = M16[row * 16 + 5 + k0];
  A2.x = M16[row * 16 + 8 + k0];  A2.y = M16[row * 16 + 9 + k0];
  A3.x = M16[row * 16 + 12 + k0]; A3.y = M16[row * 16 + 13 + k0];

  const int nTiles = NCOLS / 16;
  for (int tile = wv; tile < nTiles; tile += nWaves) {
    const int c = tile * 16 + row;
    const int base = (c & ((1 << lgL) - 1)) +
                     (((c >> lgL) & ((1 << lgM) - 1)) << (lgL + 2)) +
                     ((c >> (lgL + lgM)) << (lgL + lgM + 4));
#define PAIR_ADDR(i) (base + (((i) >> 2) * sFirst) + (((i) & 3) * sSecond))
    v2f B0, B1, B2, B3;  // B 4x16: lane holds N=row, rows K mirror A's split
    B0.x = r[PAIR_ADDR(0 + k0)];  B0.y = r[PAIR_ADDR(1 + k0)];
    B1.x = r[PAIR_ADDR(4 + k0)];  B1.y = r[PAIR_ADDR(5 + k0)];
    B2.x = r[PAIR_ADDR(8 + k0)];  B2.y = r[PAIR_ADDR(9 + k0)];
    B3.x = r[PAIR_ADDR(12 + k0)]; B3.y = r[PAIR_ADDR(13 + k0)];

    v8f C = {0.f, 0.f, 0.f, 0.f, 0.f, 0.f, 0.f, 0.f};
    C = __builtin_amdgcn_wmma_f32_16x16x4_f32(false, A0, false, B0, (short)0, C, false, false);
    C = __builtin_amdgcn_wmma_f32_16x16x4_f32(false, A1, false, B1, (short)0, C, false, false);
    C = __builtin_amdgcn_wmma_f32_16x16x4_f32(false, A2, false, B2, (short)0, C, false, false);
    C = __builtin_amdgcn_wmma_f32_16x16x4_f32(false, A3, false, B3, (short)0, C, false, false);

#pragma unroll
    for (int rr = 0; rr < 8; ++rr) {      // D: VGPR rr = row rr + 8*hi, col=row
      const int i = rr + 8 * hi;
      r[PAIR_ADDR(i)] = C[rr];
    }
#undef PAIR_ADDR
  }
}

// coef[k] = w[k] * tr(P_k rho) ; P order: k=2q -> X_q, k=2q+1 -> Z_q
__global__ void coefs(const float* __restrict__ r, const float* __restrict__ w,
                      float* __restrict__ coef) {
  const int t = threadIdx.x;
  if (t < 2 * NQ) {
    const int q = t >> 1;
    const unsigned p = (t & 1) ? 3u : 1u;         // Z : X
    coef[t] = w[t] * r[p << (2 * (9 - q))];       // qubit q digit stride 4^(9-q)
  }
}

// logits[b] = Σ_m x²·d[m] + Σ_q cX_q Σ_m x[m] x[m^mask_q] ; out = sigmoid
__global__ void __launch_bounds__(256) classify(
    const float* __restrict__ x, const float* __restrict__ bias,
    const float* __restrict__ coef, float* __restrict__ out) {
  __shared__ float xs[DIMN];
  __shared__ float red[256];
  __shared__ float cX[NQ], cZ[NQ];
  const int b = blockIdx.x, tid = threadIdx.x;
  if (tid < NQ) { cX[tid] = coef[2 * tid]; cZ[tid] = coef[2 * tid + 1]; }
  for (int m = tid; m < DIMN; m += 256) xs[m] = x[b * DIMN + m] + bias[m];
  __syncthreads();

  float acc = 0.f;
  for (int m = tid; m < DIMN; m += 256) {
    const float v = xs[m];
    float t = 0.f;
#pragma unroll
    for (int q = 0; q < NQ; ++q) {
      t += ((m >> (9 - q)) & 1) ? -cZ[q] : cZ[q];     // diagonal Z sign
    }
    t *= v;
#pragma unroll
    for (int q = 0; q < NQ; ++q) t += cX[q] * xs[m ^ (1 << (9 - q))];
    acc += v * t;
  }
  red[tid] = acc;
  __syncthreads();
  for (int s = 128; s > 0; s >>= 1) {
    if (tid < s) red[tid] += red[tid + s];
    __syncthreads();
  }
  if (tid == 0) out[b] = 1.f / (1.f + expf(-red[0]));
}

// ---------------------------------------------------------------------------
extern "C" void kernel_launch(void* const* d_in, const int* in_sizes, int n_in,
                              void* d_out, int out_size, void* d_ws, size_t ws_size,
                              hipStream_t stream) {
  const float* x = (const float*)d_in[0];     // [2048,1024]
  const float* bias = (const float*)d_in[1];  // [1024]
  const float* w = (const float*)d_in[2];     // [20]
  const float* cw = (const float*)d_in[3];    // [3,10,3]
  // d_in[4] = P_tensor: structurally known, unused.
  float* out = (float*)d_out;

  float* ws = (float*)d_ws;
  float* r = ws;                        // 4^10 floats (4 MB)
  float* mats = ws + RLEN;              // 17 x 16x16 matrices
  float* coef = mats + 17 * MAT_STRIDE; // 20 floats

  prep_mats<<<1, 32, 0, stream>>>(cw, mats);
  init_r<<<RLEN / 256, 256, 0, stream>>>(r);

  const int AB = 512;  // 512 blocks * 8 waves = 4096 waves = 4096 tiles
  for (int l = 0; l < NL; ++l) {
    // single-qubit Rot+noise on qubit pairs (commuting channels)
    for (int j = 0; j < 5; ++j) {
      const int q = 2 * j, pl = 8 - q;  // lower digit position (qubit q+1)
      apply_pair<<<AB, 256, 0, stream>>>(r, mats + (l * 5 + j) * MAT_STRIDE,
                                         2 * pl, 0, 1 << (2 * (9 - q)),
                                         1 << (2 * (8 - q)));
    }
    // CNOT chain (sequential): fused (CNOT ∘ depol2 ∘ amp ∘ phase on both)
    for (int q = 0; q < NQ; ++q) {
      if (q < 9) {
        const int pl = 8 - q;
        apply_pair<<<AB, 256, 0, stream>>>(r, mats + 15 * MAT_STRIDE,
                                           2 * pl, 0, 1 << (2 * (9 - q)),
                                           1 << (2 * (8 - q)));
      } else {
        // CNOT(9,0): control digit pos 0 (stride 1), target pos 9 (stride 4^9)
        apply_pair<<<AB, 256, 0, stream>>>(r, mats + 15 * MAT_STRIDE,
                                           0, 16, 1, 1 << 18);
      }
    }
  }
  // readout bitflip on all qubits (pairwise)
  for (int j = 0; j < 5; ++j) {
    const int q = 2 * j, pl = 8 - q;
    apply_pair<<<AB, 256, 0, stream>>>(r, mats + 16 * MAT_STRIDE,
                                       2 * pl, 0, 1 << (2 * (9 - q)),
                                       1 << (2 * (8 - q)));
  }

  coefs<<<1, 32, 0, stream>>>(r, w, coef);
  classify<<<BATCH, 256, 0, stream>>>(x, bias, coef, out);
}